// HouseholdODEFunc_79113297592762
// MI455X (gfx1250) — compile-verified
//
#include <hip/hip_runtime.h>
#include <hip/hip_bf16.h>

// ---------------------------------------------------------------------------
// MI455X (gfx1250) implementation.
//  * All GEMMs (incl. attention) run on V_WMMA_F32_16X16X4_F32 (fp32 matrix
//    pipe) -- exact precision vs the fp32 reference.
//  * Global->LDS staging uses GLOBAL_LOAD_ASYNC_TO_LDS_B128 (ASYNCcnt path,
//    no VGPR round trip) when the builtin is available; the dense GEMM
//    double-buffers its LDS tiles so async DMA overlaps the WMMA stream.
//  * Attention is flash-style: scores never touch global memory (268MB/layer
//    > L2). The q/k/v working set (~64MB) lives in the 192MB L2, so per-block
//    K/V streaming is L2 traffic, not HBM.
//  * Workspace requirement: ~206 MB (layout below).
// ---------------------------------------------------------------------------

#define NR 8192   // N rows
#define HH 512    // hidden
#define NL 3      // layers

typedef __attribute__((ext_vector_type(2))) float v2f;
typedef __attribute__((ext_vector_type(8))) float v8f;

// ---- async global->LDS copy (CDNA5) ---------------------------------------
#if defined(__has_builtin)
#if __has_builtin(__builtin_amdgcn_global_load_async_to_lds_b128) && \
    __has_builtin(__builtin_amdgcn_s_wait_asynccnt)
#define HAVE_ASYNC_LDS 1
#endif
#endif

// Builtin prototype (probe-confirmed via round-2 diagnostic):
//   (AS1 int4* gsrc, AS3 int4* ldst, imm int offset, imm int cpol)
typedef __attribute__((__vector_size__(4 * sizeof(int)))) int i4v;
typedef __attribute__((address_space(1))) i4v gbl_i4;
typedef __attribute__((address_space(3))) i4v lds_i4;

__device__ __forceinline__ void async_copy16(float* dst_lds, const float* src) {
#ifdef HAVE_ASYNC_LDS
  __builtin_amdgcn_global_load_async_to_lds_b128((gbl_i4*)src, (lds_i4*)dst_lds,
                                                 0, 0);
#else
  *(float4*)dst_lds = *(const float4*)src;
#endif
}
__device__ __forceinline__ void async_wait_all() {
#ifdef HAVE_ASYNC_LDS
  __builtin_amdgcn_s_wait_asynccnt(0);
#endif
}

__device__ __forceinline__ v8f wmma_f32(v2f a, v2f b, v8f c) {
  // D = A(16x4) * B(4x16) + C(16x16), fp32.
  return __builtin_amdgcn_wmma_f32_16x16x4_f32(false, a, false, b, (short)0, c,
                                               false, false);
}

// ---------------------------------------------------------------------------
// Generic GEMM: C[M x Nout] = act(A[M x K](row stride lda) @ W[Nout x K]^T + b) * alpha
// Block tile 128x64, 256 threads = 8 waves (4 x 2), 2x2 16x16 WMMA tiles/wave.
// Double-buffered LDS K-tiles fed by async global->LDS DMA.
// Requires: M % 128 == 0, Nout % 64 == 0 (grid.y = Nout/64), K % 16 == 0.
// ---------------------------------------------------------------------------
__global__ __launch_bounds__(256) void gemm_bias_act(
    const float* __restrict__ A, int lda, const float* __restrict__ W,
    const float* __restrict__ bias, float* __restrict__ C, int ldc, int K,
    float alpha, int act) {
  __shared__ float As[2][128 * 20];  // 16 K-cols padded to 20 (conflict-free)
  __shared__ float Bs[2][64 * 20];

  const int tid = threadIdx.x;
  const int wave = tid >> 5;
  const int lane = tid & 31;
  const int lm = lane & 15;   // row/col within 16
  const int lk = lane >> 4;   // k-half select
  const int wm = wave & 3;    // wave m-tile (32 rows each)
  const int wn = wave >> 2;   // wave n-tile (32 cols each)
  const int m0 = blockIdx.x * 128;
  const int n0 = blockIdx.y * 64;

  const int r = tid >> 2, c4 = (tid & 3) << 2;  // tile-load coordinates

  v8f acc[2][2];
  const v8f vzero = {0.f, 0.f, 0.f, 0.f, 0.f, 0.f, 0.f, 0.f};
  acc[0][0] = vzero; acc[0][1] = vzero; acc[1][0] = vzero; acc[1][1] = vzero;

  // Stage the 128x16 A tile + 64x16 W tile for column block kc into buffer b.
  auto issue_tile = [&](int kc, int b) {
    async_copy16(&As[b][r * 20 + c4], A + (size_t)(m0 + r) * lda + kc + c4);
    async_copy16(&As[b][(r + 64) * 20 + c4],
                 A + (size_t)(m0 + r + 64) * lda + kc + c4);
    async_copy16(&Bs[b][r * 20 + c4], W + (size_t)(n0 + r) * K + kc + c4);
  };

  issue_tile(0, 0);
  async_wait_all();
  __syncthreads();

  int cur = 0;
  for (int kc = 0; kc < K; kc += 16) {
    if (kc + 16 < K) issue_tile(kc + 16, cur ^ 1);  // overlap DMA with WMMA
#pragma unroll
    for (int ks = 0; ks < 16; ks += 4) {
      v2f af[2], bf[2];
#pragma unroll
      for (int i = 0; i < 2; i++)
        af[i] = *(const v2f*)&As[cur][(wm * 32 + i * 16 + lm) * 20 + ks + 2 * lk];
#pragma unroll
      for (int j = 0; j < 2; j++)
        bf[j] = *(const v2f*)&Bs[cur][(wn * 32 + j * 16 + lm) * 20 + ks + 2 * lk];
#pragma unroll
      for (int i = 0; i < 2; i++)
#pragma unroll
        for (int j = 0; j < 2; j++) acc[i][j] = wmma_f32(af[i], bf[j], acc[i][j]);
    }
    async_wait_all();
    __syncthreads();
    cur ^= 1;
  }

  // Epilogue. C/D layout: lane -> N = lm, VGPR v -> M = v + 8*lk.
#pragma unroll
  for (int i = 0; i < 2; i++)
#pragma unroll
    for (int j = 0; j < 2; j++) {
      int nn = n0 + wn * 32 + j * 16 + lm;
      float bv = bias[nn];
#pragma unroll
      for (int v = 0; v < 8; v++) {
        int mm = m0 + wm * 32 + i * 16 + v + 8 * lk;
        float x = acc[i][j][v] + bv;
        if (act) x = fmaxf(x, 0.f);
        C[(size_t)mm * ldc + nn] = x * alpha;
      }
    }
}

// ---------------------------------------------------------------------------
// Flash attention: ctx = softmax(mask_diag(q k^T / sqrt(H))) @ v
// One block = 16 query rows, 256 threads (8 waves). Streams 16-key chunks
// through LDS via async DMA. Wave w owns ctx columns [w*64, w*64+64).
// Scores: D=512 reduction split 8 ways (64 each), combined via LDS.
// ---------------------------------------------------------------------------
__global__ __launch_bounds__(256) void attention_kernel(
    const float* __restrict__ q, const float* __restrict__ k,
    const float* __restrict__ v, float* __restrict__ ctx, int ldc) {
  __shared__ float Qs[16 * 516];
  __shared__ float Ks[16 * 516];
  __shared__ float Vs[16 * 516];
  __shared__ float Sp[8 * 256];   // per-wave partial score tiles
  __shared__ float Pp[16 * 20];   // softmax probabilities (padded stride)
  __shared__ float rowm[16], rowl[16], corr[16];

  const int tid = threadIdx.x;
  const int wave = tid >> 5;
  const int lane = tid & 31;
  const int lm = lane & 15;
  const int lk = lane >> 4;
  const int q0 = blockIdx.x * 16;
  const float rs = 0.044194173824159216f;  // 1/sqrt(512)

  // Load + pre-scale Q tile (16 x 512)
  for (int s = tid; s < 16 * 128; s += 256) {
    int r = s >> 7, c4 = (s & 127) << 2;
    float4 t4 = *(const float4*)(q + (size_t)(q0 + r) * HH + c4);
    float* d = &Qs[r * 516 + c4];
    d[0] = t4.x * rs; d[1] = t4.y * rs; d[2] = t4.z * rs; d[3] = t4.w * rs;
  }
  if (tid < 16) { rowm[tid] = -INFINITY; rowl[tid] = 0.f; }

  v8f acc[4];
  const v8f vzero = {0.f, 0.f, 0.f, 0.f, 0.f, 0.f, 0.f, 0.f};
  acc[0] = vzero; acc[1] = vzero; acc[2] = vzero; acc[3] = vzero;
  __syncthreads();

  for (int j0 = 0; j0 < NR; j0 += 16) {
    // Async-stage K/V chunk (16 x 512 each) straight into LDS
#pragma unroll
    for (int it = 0; it < 8; it++) {
      int s = tid + it * 256;
      int r = s >> 7, c4 = (s & 127) << 2;
      async_copy16(&Ks[r * 516 + c4], k + (size_t)(j0 + r) * HH + c4);
      async_copy16(&Vs[r * 516 + c4], v + (size_t)(j0 + r) * HH + c4);
    }
    async_wait_all();
    __syncthreads();

    // Phase 1: partial S = Q K^T over this wave's 64-wide D slice
    v8f sa = vzero;
    const int kb = wave * 64;
#pragma unroll
    for (int kk = 0; kk < 64; kk += 4) {
      v2f af = *(const v2f*)&Qs[lm * 516 + kb + kk + 2 * lk];
      v2f bf = *(const v2f*)&Ks[lm * 516 + kb + kk + 2 * lk];
      sa = wmma_f32(af, bf, sa);
    }
#pragma unroll
    for (int vv = 0; vv < 8; vv++)
      Sp[wave * 256 + (vv + 8 * lk) * 16 + lm] = sa[vv];
    __syncthreads();

    // Phase 2a: reduce partials; mask diagonal
    {
      int m = tid >> 4, n = tid & 15;
      float s = 0.f;
#pragma unroll
      for (int w = 0; w < 8; w++) s += Sp[w * 256 + tid];
      if (q0 + m == j0 + n) s = -1e9f;
      Sp[tid] = s;
    }
    __syncthreads();
    // Phase 2b: online softmax bookkeeping (one thread per query row)
    if (tid < 16) {
      int m = tid;
      float mx = rowm[m], cmax = -INFINITY;
#pragma unroll
      for (int n = 0; n < 16; n++) cmax = fmaxf(cmax, Sp[m * 16 + n]);
      float mnew = fmaxf(mx, cmax);
      float c = __expf(mx - mnew);
      float sum = 0.f;
#pragma unroll
      for (int n = 0; n < 16; n++) {
        float p = __expf(Sp[m * 16 + n] - mnew);
        Pp[m * 20 + n] = p;
        sum += p;
      }
      rowl[m] = rowl[m] * c + sum;
      rowm[m] = mnew;
      corr[m] = c;
    }
    __syncthreads();

    // Phase 3: rescale accumulators, then ctx += P @ V (this wave's 64 cols)
    float cf[8];
#pragma unroll
    for (int vv = 0; vv < 8; vv++) cf[vv] = corr[vv + 8 * lk];
#pragma unroll
    for (int t = 0; t < 4; t++)
#pragma unroll
      for (int vv = 0; vv < 8; vv++) acc[t][vv] *= cf[vv];
#pragma unroll
    for (int ks = 0; ks < 16; ks += 4) {
      v2f af = *(const v2f*)&Pp[lm * 20 + ks + 2 * lk];
#pragma unroll
      for (int t = 0; t < 4; t++) {
        int nb = wave * 64 + t * 16 + lm;
        v2f bf;
        bf.x = Vs[(ks + 2 * lk) * 516 + nb];
        bf.y = Vs[(ks + 2 * lk + 1) * 516 + nb];
        acc[t] = wmma_f32(af, bf, acc[t]);
      }
    }
    __syncthreads();
  }

  // Epilogue: divide by softmax denominator, store ctx slice
  float linv[8];
#pragma unroll
  for (int vv = 0; vv < 8; vv++) linv[vv] = 1.f / rowl[vv + 8 * lk];
#pragma unroll
  for (int t = 0; t < 4; t++)
#pragma unroll
    for (int vv = 0; vv < 8; vv++) {
      int mm = q0 + vv + 8 * lk;
      int nn = wave * 64 + t * 16 + lm;
      ctx[(size_t)mm * ldc + nn] = acc[t][vv] * linv[vv];
    }
}

// ---------------------------------------------------------------------------
// Fused GRUCell (torch gate order r,z,n) + LayerNorm. One block per row.
// Reads/writes h in-place at hc[:, 0:512] (row stride 1024).
// ---------------------------------------------------------------------------
__global__ __launch_bounds__(256) void gru_ln_kernel(
    const float* __restrict__ gx, const float* __restrict__ gh,
    float* __restrict__ hc, const float* __restrict__ lng,
    const float* __restrict__ lnb) {
  __shared__ float red[256];
  const int n = blockIdx.x, t = threadIdx.x;
  const size_t gb = (size_t)n * 1536;
  float hm[2];
#pragma unroll
  for (int e = 0; e < 2; e++) {
    int c = t + e * 256;
    float xr = gx[gb + c] + gh[gb + c];
    float xz = gx[gb + 512 + c] + gh[gb + 512 + c];
    float r = 1.f / (1.f + __expf(-xr));
    float z = 1.f / (1.f + __expf(-xz));
    float nc = tanhf(gx[gb + 1024 + c] + r * gh[gb + 1024 + c]);
    float h = hc[(size_t)n * 1024 + c];
    hm[e] = (1.f - z) * nc + z * h;
  }
  red[t] = hm[0] + hm[1];
  __syncthreads();
  for (int off = 128; off > 0; off >>= 1) {
    if (t < off) red[t] += red[t + off];
    __syncthreads();
  }
  float mu = red[0] * (1.f / 512.f);
  __syncthreads();
  float d0 = hm[0] - mu, d1 = hm[1] - mu;
  red[t] = d0 * d0 + d1 * d1;
  __syncthreads();
  for (int off = 128; off > 0; off >>= 1) {
    if (t < off) red[t] += red[t + off];
    __syncthreads();
  }
  float inv = rsqrtf(red[0] * (1.f / 512.f) + 1e-5f);
#pragma unroll
  for (int e = 0; e < 2; e++) {
    int c = t + e * 256;
    hc[(size_t)n * 1024 + c] = (hm[e] - mu) * inv * lng[c] + lnb[c];
  }
}

// ---------------------------------------------------------------------------
// Small prep kernels
// ---------------------------------------------------------------------------
__global__ void te_kernel(const float* __restrict__ t,
                          const float* __restrict__ Wt,
                          const float* __restrict__ bt, float* __restrict__ te) {
  int i = threadIdx.x;  // 128
  te[i] = t[0] * Wt[i] + bt[i];
}

__global__ void fill_x_kernel(const float* __restrict__ state,
                              const float* __restrict__ te,
                              float* __restrict__ x) {
  size_t idx = (size_t)blockIdx.x * 256 + threadIdx.x;  // N*512
  int c = (int)(idx & 511);
  size_t n = idx >> 9;
  x[idx] = (c < 384) ? state[n * 512 + c] : te[c - 384];
}

__global__ void fill_ig_kernel(const float* __restrict__ te,
                               float* __restrict__ ig) {
  size_t idx = (size_t)blockIdx.x * 256 + threadIdx.x;  // N*128
  int c = (int)(idx & 127);
  size_t n = idx >> 7;
  ig[n * 640 + 512 + c] = te[c];
}

// ---------------------------------------------------------------------------
extern "C" void kernel_launch(void* const* d_in, const int* in_sizes, int n_in,
                              void* d_out, int out_size, void* d_ws,
                              size_t ws_size, hipStream_t stream) {
  (void)in_sizes; (void)n_in; (void)out_size; (void)ws_size;
  const float* t      = (const float*)d_in[0];
  const float* state  = (const float*)d_in[1];
  const float* W_time = (const float*)d_in[2];
  const float* b_time = (const float*)d_in[3];
  const float* W_in   = (const float*)d_in[4];
  const float* b_in   = (const float*)d_in[5];
  const float* Wq     = (const float*)d_in[6];
  const float* bq     = (const float*)d_in[7];
  const float* Wk     = (const float*)d_in[8];
  const float* bk     = (const float*)d_in[9];
  const float* Wv     = (const float*)d_in[10];
  const float* bv     = (const float*)d_in[11];
  const float* Wint   = (const float*)d_in[12];
  const float* bint   = (const float*)d_in[13];
  const float* W_ih   = (const float*)d_in[14];
  const float* b_ih   = (const float*)d_in[15];
  const float* W_hh   = (const float*)d_in[16];
  const float* b_hh   = (const float*)d_in[17];
  const float* ln_g   = (const float*)d_in[18];
  const float* ln_b   = (const float*)d_in[19];
  const float* W_out  = (const float*)d_in[20];
  const float* b_out  = (const float*)d_in[21];
  float* out = (float*)d_out;

  // Workspace layout (floats): total 256 + N*(1024+512*3+640+1536*2) ~= 206MB
  float* ws = (float*)d_ws;
  float* te = ws;                           // 128 (padded to 256)
  float* hc = te + 256;                     // [N,1024] = [h | ctx]
  float* q  = hc + (size_t)NR * 1024;       // [N,512]
  float* kx = q  + (size_t)NR * 512;        // [N,512]
  float* vx = kx + (size_t)NR * 512;        // [N,512]
  float* ig = vx + (size_t)NR * 512;        // [N,640] = [inter | te]
  float* gx = ig + (size_t)NR * 640;        // [N,1536] (also aliased as x [N,512])
  float* gh = gx + (size_t)NR * 1536;       // [N,1536]

  const dim3 blk(256);
  const dim3 g512(NR / 128, 512 / 64);
  const dim3 g1536(NR / 128, 1536 / 64);

  te_kernel<<<1, 128, 0, stream>>>(t, W_time, b_time, te);
  fill_x_kernel<<<(NR * 512) / 256, blk, 0, stream>>>(state, te, gx);
  fill_ig_kernel<<<(NR * 128) / 256, blk, 0, stream>>>(te, ig);

  // h = relu(x @ W_in^T + b_in) -> hc[:, 0:512]
  gemm_bias_act<<<g512, blk, 0, stream>>>(gx, 512, W_in, b_in, hc, 1024, 512,
                                          1.f, 1);
  for (int l = 0; l < NL; l++) {
    gemm_bias_act<<<g512, blk, 0, stream>>>(hc, 1024, Wq + (size_t)l * HH * HH,
                                            bq + l * HH, q, 512, 512, 1.f, 0);
    gemm_bias_act<<<g512, blk, 0, stream>>>(hc, 1024, Wk + (size_t)l * HH * HH,
                                            bk + l * HH, kx, 512, 512, 1.f, 0);
    gemm_bias_act<<<g512, blk, 0, stream>>>(hc, 1024, Wv + (size_t)l * HH * HH,
                                            bv + l * HH, vx, 512, 512, 1.f, 0);
    attention_kernel<<<NR / 16, blk, 0, stream>>>(q, kx, vx, hc + 512, 1024);
    // inter = relu([h|ctx] @ Wint^T + bint) -> ig[:, 0:512]
    gemm_bias_act<<<g512, blk, 0, stream>>>(hc, 1024,
                                            Wint + (size_t)l * HH * 1024,
                                            bint + l * HH, ig, 640, 1024, 1.f, 1);
    // gx = [inter|te] @ W_ih^T + b_ih
    gemm_bias_act<<<g1536, blk, 0, stream>>>(ig, 640,
                                             W_ih + (size_t)l * 1536 * 640,
                                             b_ih + l * 1536, gx, 1536, 640,
                                             1.f, 0);
    // gh = h @ W_hh^T + b_hh
    gemm_bias_act<<<g1536, blk, 0, stream>>>(hc, 1024,
                                             W_hh + (size_t)l * 1536 * 512,
                                             b_hh + l * 1536, gh, 1536, 512,
                                             1.f, 0);
    gru_ln_kernel<<<NR, blk, 0, stream>>>(gx, gh, hc, ln_g + l * 512,
                                          ln_b + l * 512);
  }
  // out = 0.1 * (h @ W_out^T + b_out)
  gemm_bias_act<<<g512, blk, 0, stream>>>(hc, 1024, W_out, b_out, out, 512, 512,
                                          0.1f, 0);
}